// Co_GCN_31258771980806
// MI455X (gfx1250) — compile-verified
//
#include <hip/hip_runtime.h>
#include <stdint.h>

#define GK 4
#define GN 8192
#define F_IN 128
#define F_HID 64
#define F_OUT 16

typedef float v2f __attribute__((ext_vector_type(2)));
typedef float v8f __attribute__((ext_vector_type(8)));

// ---------------------------------------------------------------------------
// JAX threefry2x32 dropout mask: jax.random.bernoulli(key(42), 0.7, (8192,64))
// ---------------------------------------------------------------------------
__device__ __forceinline__ uint32_t rotl32(uint32_t x, uint32_t r) {
  return (x << r) | (x >> (32u - r));
}

__device__ __forceinline__ bool dropout_keep(uint32_t idx) {
  const uint32_t ks0 = 0u, ks1 = 42u;
  const uint32_t ks2 = 0x1BD11BDAu ^ ks0 ^ ks1;
  const uint32_t halfn = (GN * F_HID) / 2;  // 262144
  const bool second = idx >= halfn;
  uint32_t c0 = second ? idx - halfn : idx;
  uint32_t c1 = second ? idx : idx + halfn;
  uint32_t x0 = c0 + ks0, x1 = c1 + ks1;
#define TF4(a, b, c, d)                         \
  x0 += x1; x1 = rotl32(x1, a); x1 ^= x0;       \
  x0 += x1; x1 = rotl32(x1, b); x1 ^= x0;       \
  x0 += x1; x1 = rotl32(x1, c); x1 ^= x0;       \
  x0 += x1; x1 = rotl32(x1, d); x1 ^= x0;
  TF4(13, 15, 26, 6)  x0 += ks1; x1 += ks2 + 1u;
  TF4(17, 29, 16, 24) x0 += ks2; x1 += ks0 + 2u;
  TF4(13, 15, 26, 6)  x0 += ks0; x1 += ks1 + 3u;
  TF4(17, 29, 16, 24) x0 += ks1; x1 += ks2 + 4u;
  TF4(13, 15, 26, 6)  x0 += ks2; x1 += ks0 + 5u;
#undef TF4
  uint32_t bits = second ? x1 : x0;
  float u = __uint_as_float((bits >> 9) | 0x3f800000u) - 1.0f;
  return u < 0.7f;  // keep-probability = 1 - 0.3
}

// ---------------------------------------------------------------------------
// softmax over 4 pi values
// ---------------------------------------------------------------------------
__global__ void softmax4_kernel(const float* __restrict__ pi, float* __restrict__ o) {
  if (threadIdx.x == 0) {
    float m = fmaxf(fmaxf(pi[0], pi[1]), fmaxf(pi[2], pi[3]));
    float e0 = __expf(pi[0] - m), e1 = __expf(pi[1] - m);
    float e2 = __expf(pi[2] - m), e3 = __expf(pi[3] - m);
    float inv = 1.0f / (e0 + e1 + e2 + e3);
    o[0] = e0 * inv; o[1] = e1 * inv; o[2] = e2 * inv; o[3] = e3 * inv;
  }
}

// ---------------------------------------------------------------------------
// Small feature GEMM: Ct[FD,MD] = (X[MD,KD] @ W[KD,FD])^T, one wave per 16x16
// tile, fp32 WMMA 16x16x4. All shapes are compile-time so the W-row pair load
// becomes base + immediate offsets with simple pointer marching. Output is
// written TRANSPOSED (column-major support) so the big adjacency kernels can
// read B operands as contiguous float2.
// A layout: lane%16 = M row, lane/16 selects K pair {0,1}/{2,3} in 2 VGPRs.
// B layout mirrors A with lane%16 = N col. C: vgpr rr -> M = rr+8*(lane/16).
// ---------------------------------------------------------------------------
template <int KD, int FD, int MD>
__global__ void wmma_gemm16_t_kernel(const float* __restrict__ X,
                                     const float* __restrict__ W,
                                     float* __restrict__ Ct) {
  constexpr int TILES_F = FD / 16;
  int wid = (int)(blockIdx.x * blockDim.x + threadIdx.x) >> 5;
  int lane = threadIdx.x & 31;
  int kh = lane >> 4;       // K-half select
  int r  = lane & 15;       // row (A) / col (B)
  int m0 = (wid / TILES_F) << 4;
  int f0 = (wid % TILES_F) << 4;

  v8f acc = {0.f, 0.f, 0.f, 0.f, 0.f, 0.f, 0.f, 0.f};
  const float* xrow = X + (size_t)(m0 + r) * KD + 2 * kh;
  const float* wcol = W + (size_t)(2 * kh) * FD + f0 + r;
#pragma unroll 4
  for (int kk = 0; kk < KD; kk += 4) {
    v2f a = *(const v2f*)(xrow + kk);
    v2f b;
    b.x = wcol[(size_t)kk * FD];
    b.y = wcol[(size_t)kk * FD + FD];
    acc = __builtin_amdgcn_wmma_f32_16x16x4_f32(false, a, false, b, (short)0, acc,
                                                false, false);
  }
#pragma unroll
  for (int rr = 0; rr < 8; ++rr) {
    // transposed store: Ct[col][row]
    Ct[(size_t)(f0 + r) * MD + (m0 + rr + 8 * kh)] = acc[rr];
  }
}

// ---------------------------------------------------------------------------
// Fused mixed-adjacency layer:  out[16 x F] block = (sum_k pi[k]*adj[k]) @ B + bias
// B supplied TRANSPOSED as Bt[F][GN] so each lane's K-pair is one b64 load at a
// compile-time immediate offset. One WG (8 waves) per 16-row block; split-K=8
// across waves; fp32 WMMA 16x16x4; deterministic LDS reduction; optional fused
// ReLU + JAX dropout (layer 1).
// ---------------------------------------------------------------------------
template <int NT, bool L1>
__global__ __launch_bounds__(256) void alayer_kernel(
    const float* __restrict__ adj, const float* __restrict__ piv,
    const float* __restrict__ Bt, const float* __restrict__ bias,
    float* __restrict__ out) {
  constexpr int F = NT * 16;
  __shared__ float red[8][16 * F];

  const int wave = threadIdx.x >> 5;
  const int lane = threadIdx.x & 31;
  const int kh = lane >> 4;
  const int r  = lane & 15;
  const int m0 = blockIdx.x << 4;

  const float p0 = piv[0], p1 = piv[1], p2 = piv[2], p3 = piv[3];
  const size_t plane = (size_t)GN * GN;
  const float* a0 = adj + (size_t)(m0 + r) * GN + 2 * kh;   // A row base
  const float* bc = Bt + (size_t)r * GN + 2 * kh;           // B column base

  v8f acc[NT];
#pragma unroll
  for (int t = 0; t < NT; ++t) acc[t] = (v8f){0.f, 0.f, 0.f, 0.f, 0.f, 0.f, 0.f, 0.f};

  const int kstart = wave * (GN / 8);
  const int kend = kstart + (GN / 8);
#pragma unroll 2
  for (int k = kstart; k < kend; k += 4) {
    // mix 4 adjacency planes with pi weights (streamed once from HBM)
    v2f q0 = *(const v2f*)(a0 + k);
    v2f q1 = *(const v2f*)(a0 + plane + k);
    v2f q2 = *(const v2f*)(a0 + 2 * plane + k);
    v2f q3 = *(const v2f*)(a0 + 3 * plane + k);
    v2f a = q0 * p0 + q1 * p1 + q2 * p2 + q3 * p3;  // packed f32 FMA

    const float* bk = bc + k;
#pragma unroll
    for (int t = 0; t < NT; ++t) {
      // contiguous K-pair for this lane's column; t offset is a compile-time
      // immediate (t * 16 rows * GN * 4B <= 1.5 MB < 24-bit ioffset range)
      v2f b = *(const v2f*)(bk + (size_t)t * 16 * GN);
      acc[t] = __builtin_amdgcn_wmma_f32_16x16x4_f32(false, a, false, b, (short)0,
                                                     acc[t], false, false);
    }
  }

  // dump per-wave partial tiles to LDS (C layout: vgpr rr -> row rr+8*kh, col=16t+r)
#pragma unroll
  for (int t = 0; t < NT; ++t) {
#pragma unroll
    for (int rr = 0; rr < 8; ++rr) {
      red[wave][(rr + 8 * kh) * F + t * 16 + r] = acc[t][rr];
    }
  }
  __syncthreads();

  // fixed-order split-K reduction + epilogue (deterministic)
  for (int e = threadIdx.x; e < 16 * F; e += 256) {
    float s = 0.f;
#pragma unroll
    for (int w = 0; w < 8; ++w) s += red[w][e];
    int row = e / F, col = e % F;
    s += bias[col];
    if (L1) {
      s = fmaxf(s, 0.f);  // ReLU
      uint32_t idx = (uint32_t)(m0 + row) * F_HID + (uint32_t)col;
      s = dropout_keep(idx) ? s * (1.0f / 0.7f) : 0.0f;
    }
    out[(size_t)(m0 + row) * F + col] = s;
  }
}

// ---------------------------------------------------------------------------
extern "C" void kernel_launch(void* const* d_in, const int* in_sizes, int n_in,
                              void* d_out, int out_size, void* d_ws, size_t ws_size,
                              hipStream_t stream) {
  const float* adj = (const float*)d_in[0];
  const float* x   = (const float*)d_in[1];
  const float* W1  = (const float*)d_in[2];
  const float* b1  = (const float*)d_in[3];
  const float* W2  = (const float*)d_in[4];
  const float* b2  = (const float*)d_in[5];
  const float* pi1 = (const float*)d_in[6];
  const float* pi2 = (const float*)d_in[7];
  float* out = (float*)d_out;

  float* ws = (float*)d_ws;
  float* pi1_n     = ws;                          // 4 floats (padded to 64)
  float* support1t = ws + 64;                     // [64][8192]
  float* h         = support1t + GN * F_HID;      // [8192][64] row-major
  float* support2t = h + GN * F_HID;              // [16][8192]

  // 1) softmax(pi1)
  softmax4_kernel<<<1, 32, 0, stream>>>(pi1, pi1_n);

  // 2) support1^T = (x @ W1)^T   (512*4 = 2048 tiles, 8 waves/block -> 256 blocks)
  wmma_gemm16_t_kernel<F_IN, F_HID, GN><<<256, 256, 0, stream>>>(x, W1, support1t);

  // 3) h = dropout(relu(A1 @ support1 + b1))   (streams adj once: 1 GiB)
  alayer_kernel<4, true><<<GN / 16, 256, 0, stream>>>(adj, pi1_n, support1t, b1, h);

  // 4) support2^T = (h @ W2)^T   (512 tiles -> 64 blocks * 8 waves)
  wmma_gemm16_t_kernel<F_HID, F_OUT, GN><<<64, 256, 0, stream>>>(h, W2, support2t);

  // 5) out = A2 @ support2 + b2   (streams adj a second time: 1 GiB)
  alayer_kernel<1, false><<<GN / 16, 256, 0, stream>>>(adj, pi2, support2t, b2, out);
}